// SpatialTransformer_69750268887316
// MI455X (gfx1250) — compile-verified
//
#include <hip/hip_runtime.h>
#include <hip/hip_bf16.h>

// Spatial transformer (trilinear grid-sample) for x:[1,4,128,128,128],
// flow:[1,3,128,128,128]. HBM-bound gather kernel; CDNA5 async-to-LDS
// pipeline stages the streaming flow reads, gathers ride the 192MB L2.
// Branch-free sampling: out-of-bounds corners are clamped and their
// per-axis weight zeroed (== reference's clip + where(valid, w, 0)).

#define TPB     256
#define NSIDE   128
#define HW      (128 * 128)
#define DHW     (128 * 128 * 128)
#define NTILES  (DHW / TPB)     // 8192
#define NBLOCKS 2048

#if defined(__HIP_DEVICE_COMPILE__) && __has_builtin(__builtin_amdgcn_global_load_async_to_lds_b32)
#define USE_ASYNC_LDS 1
#else
#define USE_ASYNC_LDS 0
#endif

#if USE_ASYNC_LDS
typedef __attribute__((address_space(1))) int* gptr_t;
typedef __attribute__((address_space(3))) int* lptr_t;
__device__ __forceinline__ void async_flow_ld(const float* g, float* l) {
  __builtin_amdgcn_global_load_async_to_lds_b32(
      (gptr_t)(g), (lptr_t)(l), /*offset=*/0, /*cpol=*/0);
}
__device__ __forceinline__ void wait_async_le3() {
#if __has_builtin(__builtin_amdgcn_s_wait_asynccnt)
  __builtin_amdgcn_s_wait_asynccnt(3);
#else
  asm volatile("s_wait_asynccnt 0x3" ::: "memory");
#endif
}
__device__ __forceinline__ void wait_async_0() {
#if __has_builtin(__builtin_amdgcn_s_wait_asynccnt)
  __builtin_amdgcn_s_wait_asynccnt(0);
#else
  asm volatile("s_wait_asynccnt 0x0" ::: "memory");
#endif
}
#endif

__device__ __forceinline__ int clamp127(int v) {
  return v < 0 ? 0 : (v > 127 ? 127 : v);   // lowers to v_med3_i32
}

__device__ __forceinline__ void process_voxel(
    const float* __restrict__ x, float* __restrict__ out,
    int vox, float fd, float fh, float fw)
{
  const int wI = vox & 127;
  const int hI = (vox >> 7) & 127;
  const int dI = vox >> 14;

  // Reference math, same op order: g = ((idx+f)/127 - 0.5)*2 ; i = ((g+1)*128 - 1)*0.5
  float gx = ((float)wI + fw) / 127.0f; gx = (gx - 0.5f) * 2.0f;
  float gy = ((float)hI + fh) / 127.0f; gy = (gy - 0.5f) * 2.0f;
  float gz = ((float)dI + fd) / 127.0f; gz = (gz - 0.5f) * 2.0f;
  const float ix = ((gx + 1.0f) * 128.0f - 1.0f) * 0.5f;
  const float iy = ((gy + 1.0f) * 128.0f - 1.0f) * 0.5f;
  const float iz = ((gz + 1.0f) * 128.0f - 1.0f) * 0.5f;

  const float xf = floorf(ix), yf = floorf(iy), zf = floorf(iz);
  const float tx = ix - xf, ty = iy - yf, tz = iz - zf;
  const int x0 = (int)xf, y0 = (int)yf, z0 = (int)zf;

  // Per-axis weights with validity folded in: invalid corner coordinate
  // -> that axis weight = 0, so the corner's product weight is exactly 0.
  const float wx0 = ((unsigned)x0 < 128u)       ? (1.0f - tx) : 0.0f;
  const float wx1 = ((unsigned)(x0 + 1) < 128u) ? tx          : 0.0f;
  const float wy0 = ((unsigned)y0 < 128u)       ? (1.0f - ty) : 0.0f;
  const float wy1 = ((unsigned)(y0 + 1) < 128u) ? ty          : 0.0f;
  const float wz0 = ((unsigned)z0 < 128u)       ? (1.0f - tz) : 0.0f;
  const float wz1 = ((unsigned)(z0 + 1) < 128u) ? tz          : 0.0f;

  // Corner weights, same association as reference: (wz*wy)*wx.
  const float c00 = wz0 * wy0, c01 = wz0 * wy1;
  const float c10 = wz1 * wy0, c11 = wz1 * wy1;
  const float w000 = c00 * wx0, w001 = c00 * wx1;
  const float w010 = c01 * wx0, w011 = c01 * wx1;
  const float w100 = c10 * wx0, w101 = c10 * wx1;
  const float w110 = c11 * wx0, w111 = c11 * wx1;

  // Clamped corner addresses (reference also gathers clamped indices).
  const int xc0 = clamp127(x0), xc1 = clamp127(x0 + 1);
  const int yb0 = clamp127(y0) << 7, yb1 = clamp127(y0 + 1) << 7;
  const int zb0 = clamp127(z0) << 14, zb1 = clamp127(z0 + 1) << 14;

  const int i000 = zb0 + yb0 + xc0, i001 = zb0 + yb0 + xc1;
  const int i010 = zb0 + yb1 + xc0, i011 = zb0 + yb1 + xc1;
  const int i100 = zb1 + yb0 + xc0, i101 = zb1 + yb0 + xc1;
  const int i110 = zb1 + yb1 + xc0, i111 = zb1 + yb1 + xc1;

  float acc[4];
#pragma unroll
  for (int c = 0; c < 4; ++c) {
    const float* p = x + c * DHW;
    // Accumulate in reference (dz,dy,dx) corner order.
    float a = p[i000] * w000;
    a = fmaf(p[i001], w001, a);
    a = fmaf(p[i010], w010, a);
    a = fmaf(p[i011], w011, a);
    a = fmaf(p[i100], w100, a);
    a = fmaf(p[i101], w101, a);
    a = fmaf(p[i110], w110, a);
    a = fmaf(p[i111], w111, a);
    acc[c] = a;
  }

  // Write-once output: non-temporal, keep L2 resident for x gathers.
  __builtin_nontemporal_store(acc[0], out + vox);
  __builtin_nontemporal_store(acc[1], out + DHW + vox);
  __builtin_nontemporal_store(acc[2], out + 2 * DHW + vox);
  __builtin_nontemporal_store(acc[3], out + 3 * DHW + vox);
}

__global__ __launch_bounds__(TPB) void SpatialTransformer_gfx1250_kernel(
    const float* __restrict__ x,
    const float* __restrict__ flow,
    float* __restrict__ out)
{
  const int tid = threadIdx.x;

#if USE_ASYNC_LDS
  __shared__ float sflow[2][3][TPB];
  const int stride = gridDim.x;
  int tile = blockIdx.x;

  // Prologue: stage first tile's flow into LDS asynchronously.
  {
    const int vox = tile * TPB + tid;
    async_flow_ld(flow + vox,           &sflow[0][0][tid]);
    async_flow_ld(flow + DHW + vox,     &sflow[0][1][tid]);
    async_flow_ld(flow + 2 * DHW + vox, &sflow[0][2][tid]);
  }
  int buf = 0;
  while (tile < NTILES) {
    const int next = tile + stride;
    if (next < NTILES) {
      // Issue next tile's 3 async loads, then wait until only those 3
      // remain outstanding -> current buffer is complete in LDS.
      const int nv = next * TPB + tid;
      async_flow_ld(flow + nv,           &sflow[buf ^ 1][0][tid]);
      async_flow_ld(flow + DHW + nv,     &sflow[buf ^ 1][1][tid]);
      async_flow_ld(flow + 2 * DHW + nv, &sflow[buf ^ 1][2][tid]);
      wait_async_le3();
    } else {
      wait_async_0();
    }
    asm volatile("" ::: "memory");  // don't hoist LDS reads above the wait
    const float fd = sflow[buf][0][tid];
    const float fh = sflow[buf][1][tid];
    const float fw = sflow[buf][2][tid];
    process_voxel(x, out, tile * TPB + tid, fd, fh, fw);
    buf ^= 1;
    tile = next;
  }
#else
  for (int tile = blockIdx.x; tile < NTILES; tile += gridDim.x) {
    const int vox = tile * TPB + tid;
    const int nvox = vox + (int)(gridDim.x) * TPB;
    if (nvox < DHW) {
      __builtin_prefetch(flow + nvox, 0, 1);
      __builtin_prefetch(flow + DHW + nvox, 0, 1);
      __builtin_prefetch(flow + 2 * DHW + nvox, 0, 1);
    }
    const float fd = flow[vox];
    const float fh = flow[DHW + vox];
    const float fw = flow[2 * DHW + vox];
    process_voxel(x, out, vox, fd, fh, fw);
  }
#endif
}

extern "C" void kernel_launch(void* const* d_in, const int* in_sizes, int n_in,
                              void* d_out, int out_size, void* d_ws, size_t ws_size,
                              hipStream_t stream) {
  (void)in_sizes; (void)n_in; (void)out_size; (void)d_ws; (void)ws_size;
  const float* x    = (const float*)d_in[0];   // [1,4,128,128,128] f32
  const float* flow = (const float*)d_in[1];   // [1,3,128,128,128] f32
  float* out = (float*)d_out;                  // [1,4,128,128,128] f32
  SpatialTransformer_gfx1250_kernel<<<dim3(NBLOCKS), dim3(TPB), 0, stream>>>(x, flow, out);
}